// GAT_B_50448685859074
// MI455X (gfx1250) — compile-verified
//
#include <hip/hip_runtime.h>
#include <hip/hip_bf16.h>
#include <math.h>

// ---------------------------------------------------------------------------
// 2-layer GAT forward for MI455X (gfx1250, wave32).
//   GEMMs via V_WMMA_F32_16X16X4_F32, one wave -> 16x64 output strip
//   (4 independent accumulator chains, A-fragment reused 4x, software-
//   pipelined loads so WMMAs wait only on the *previous* iteration's loads).
//   Edge softmax/aggregation via L2-resident f32 atomics (all per-node state
//   ~20MB/buffer fits the 192MB L2; HBM sees only streaming passes).
// ---------------------------------------------------------------------------

typedef __attribute__((ext_vector_type(2))) float v2f;
typedef __attribute__((ext_vector_type(8))) float v8f;

// ---------------- generic fill ----------------
__global__ void fill_f32(float* __restrict__ p, float v, int n) {
    int i = blockIdx.x * blockDim.x + threadIdx.x;
    if (i < n) p[i] = v;
}

// ---------------- degree ----------------
__global__ void degree_k(const int* __restrict__ dst, float* __restrict__ deg, int E) {
    int i = blockIdx.x * blockDim.x + threadIdx.x;
    if (i < E) atomicAdd(&deg[dst[i]], 1.0f);
}

// ---------------- GEMM: C[M,256] = A[M,K] @ W[256,K]^T  (opt +=) ----------
// One wave -> 16 rows x 64 cols (4 WMMA tiles). K stepped by 4 via
// V_WMMA_F32_16X16X4_F32.  fp32 A-frag 16x4 layout: lanes 0-15 hold
// K={k,k+1}, lanes 16-31 hold K={k+2,k+3}; B-frag mirrors with cols across
// lanes, B[k][n] = W[n][k].  Inner loop is software-pipelined: fragments for
// step k+4 are fetched before the 4 WMMAs of step k are issued.
__global__ void gemm_wmma_f32(const float* __restrict__ A, int lda,
                              const float* __restrict__ W, int ldw,
                              float* __restrict__ C, int ldc,
                              int M, int K, int accumulate)
{
    const int nColGroups = 4;                       // 4 groups of 64 cols = 256
    int waveId = blockIdx.x * (blockDim.x >> 5) + (threadIdx.x >> 5);
    int lane   = threadIdx.x & 31;
    int nRowTiles = M >> 4;
    if (waveId >= nRowTiles * nColGroups) return;   // wave-uniform: EXEC all-1
    int rowTile  = waveId / nColGroups;
    int colGroup = waveId - rowTile * nColGroups;
    int col0     = colGroup * 64;
    int half = lane >> 4;        // 0: lanes 0-15, 1: lanes 16-31
    int lr   = lane & 15;

    const float* Arow = A + (size_t)(rowTile * 16 + lr) * lda + 2 * half;
    const float* Wr0  = W + (size_t)(col0 +  0 + lr) * ldw + 2 * half;
    const float* Wr1  = W + (size_t)(col0 + 16 + lr) * ldw + 2 * half;
    const float* Wr2  = W + (size_t)(col0 + 32 + lr) * ldw + 2 * half;
    const float* Wr3  = W + (size_t)(col0 + 48 + lr) * ldw + 2 * half;

    v8f acc0 = {}, acc1 = {}, acc2 = {}, acc3 = {};

    // prologue: fetch k=0 fragments (8B-aligned v2f loads)
    v2f a  = *(const v2f*)(Arow);
    v2f b0 = *(const v2f*)(Wr0);
    v2f b1 = *(const v2f*)(Wr1);
    v2f b2 = *(const v2f*)(Wr2);
    v2f b3 = *(const v2f*)(Wr3);

#pragma unroll 2
    for (int k = 4; k < K; k += 4) {
        // fetch next iteration's fragments first (hides L2 latency behind WMMA)
        v2f an  = *(const v2f*)(Arow + k);
        v2f bn0 = *(const v2f*)(Wr0 + k);
        v2f bn1 = *(const v2f*)(Wr1 + k);
        v2f bn2 = *(const v2f*)(Wr2 + k);
        v2f bn3 = *(const v2f*)(Wr3 + k);
        acc0 = __builtin_amdgcn_wmma_f32_16x16x4_f32(false, a, false, b0, (short)0, acc0, false, false);
        acc1 = __builtin_amdgcn_wmma_f32_16x16x4_f32(false, a, false, b1, (short)0, acc1, false, false);
        acc2 = __builtin_amdgcn_wmma_f32_16x16x4_f32(false, a, false, b2, (short)0, acc2, false, false);
        acc3 = __builtin_amdgcn_wmma_f32_16x16x4_f32(false, a, false, b3, (short)0, acc3, false, false);
        a = an; b0 = bn0; b1 = bn1; b2 = bn2; b3 = bn3;
    }
    // epilogue
    acc0 = __builtin_amdgcn_wmma_f32_16x16x4_f32(false, a, false, b0, (short)0, acc0, false, false);
    acc1 = __builtin_amdgcn_wmma_f32_16x16x4_f32(false, a, false, b1, (short)0, acc1, false, false);
    acc2 = __builtin_amdgcn_wmma_f32_16x16x4_f32(false, a, false, b2, (short)0, acc2, false, false);
    acc3 = __builtin_amdgcn_wmma_f32_16x16x4_f32(false, a, false, b3, (short)0, acc3, false, false);

    // C/D layout: VGPR j -> row (8*half + j), col (lane&15) within each tile
    float* Cb = C + (size_t)(rowTile * 16 + 8 * half) * ldc + col0 + lr;
#pragma unroll
    for (int j = 0; j < 8; ++j) {
        float* p = Cb + (size_t)j * ldc;
        p[ 0] = (accumulate ? p[ 0] : 0.0f) + acc0[j];
        p[16] = (accumulate ? p[16] : 0.0f) + acc1[j];
        p[32] = (accumulate ? p[32] : 0.0f) + acc2[j];
        p[48] = (accumulate ? p[48] : 0.0f) + acc3[j];
    }
}

// ---------------- attention scores: el/er per (node,head) -----------------
// One wave per (n,h) pair; feat pair-row is contiguous D floats.
__global__ void attn_scores(const float* __restrict__ feat,
                            const float* __restrict__ attn_l,
                            const float* __restrict__ attn_r,
                            float* __restrict__ el, float* __restrict__ er,
                            int pairs, int H, int D)
{
    int p    = blockIdx.x * (blockDim.x >> 5) + (threadIdx.x >> 5);
    int lane = threadIdx.x & 31;
    if (p >= pairs) return;
    int h = p % H;
    const float* f  = feat + (size_t)p * D;
    const float* al = attn_l + (size_t)h * D;
    const float* ar = attn_r + (size_t)h * D;
    float sl = 0.f, sr = 0.f;
    for (int i = lane; i < D; i += 32) {
        float fv = f[i];
        sl += fv * al[i];
        sr += fv * ar[i];
    }
#pragma unroll
    for (int off = 16; off > 0; off >>= 1) {
        sl += __shfl_xor(sl, off, 32);
        sr += __shfl_xor(sr, off, 32);
    }
    if (lane == 0) { el[p] = sl; er[p] = sr; }
}

// ---------------- float atomic max via ordered int trick ------------------
__device__ inline void atomicMaxF(float* addr, float val) {
    if (val >= 0.0f) atomicMax((int*)addr, __float_as_int(val));
    else             atomicMin((unsigned int*)addr, __float_as_uint(val));
}

__device__ inline float leaky02(float v) { return v > 0.0f ? v : 0.2f * v; }

// ---------------- edge pass A: segment max of leaky(el[src]+er[dst]) -------
// NOTE: no -inf fixup pass is needed afterwards: m[dst*H+h] is only ever read
// at destinations that received >=1 edge here, where it is finite.
__global__ void edge_max_k(const int* __restrict__ src, const int* __restrict__ dst,
                           const float* __restrict__ el, const float* __restrict__ er,
                           float* __restrict__ m, int E, int H)
{
    int t = blockIdx.x * blockDim.x + threadIdx.x;
    if (t >= E * H) return;
    int e = t / H, h = t - e * H;
    float v = leaky02(el[src[e] * H + h] + er[dst[e] * H + h]);
    atomicMaxF(&m[dst[e] * H + h], v);
}

// ---------------- edge pass B: ex = exp(e - m[dst]); segment sum ----------
__global__ void edge_expsum_k(const int* __restrict__ src, const int* __restrict__ dst,
                              const float* __restrict__ el, const float* __restrict__ er,
                              const float* __restrict__ m, float* __restrict__ s,
                              float* __restrict__ exb, int E, int H)
{
    int t = blockIdx.x * blockDim.x + threadIdx.x;
    if (t >= E * H) return;
    int e = t / H, h = t - e * H;
    int d = dst[e];
    float v  = leaky02(el[src[e] * H + h] + er[d * H + h]);
    float ex = expf(v - m[d * H + h]);
    exb[t] = ex;
    atomicAdd(&s[d * H + h], ex);
}

// ---------------- edge pass C: agg[dst] += feat[src] * a ------------------
// thread handles 4 consecutive feature dims of one edge (float4 gather,
// 4 atomic f32 adds; destination array is L2-resident).
__global__ void edge_agg_k(const int* __restrict__ src, const int* __restrict__ dst,
                           const float* __restrict__ exb, const float* __restrict__ s,
                           const float* __restrict__ feat, float* __restrict__ agg,
                           int total, int H, int D, int F)
{
    int t = blockIdx.x * blockDim.x + threadIdx.x;
    if (t >= total) return;
    int chunks = F >> 2;            // float4 chunks per edge
    int e  = t / chunks;
    int q  = t - e * chunks;
    int d0 = q << 2;
    int h  = d0 / D;                // 4 consecutive dims never straddle a head
    int sN = src[e], dN = dst[e];
    float a = exb[e * H + h] / s[dN * H + h];
    const float4 f = *(const float4*)(feat + (size_t)sN * F + d0);
    float* o = agg + (size_t)dN * F + d0;
    atomicAdd(o + 0, f.x * a);
    atomicAdd(o + 1, f.y * a);
    atomicAdd(o + 2, f.z * a);
    atomicAdd(o + 3, f.w * a);
}

// ---------------- layer-1 combine: h1 = relu(agg + b1 + norm*(buf+bb0)) ---
__global__ void combine_h1_k(const float* __restrict__ agg, const float* __restrict__ bias1,
                             const float* __restrict__ buf, const float* __restrict__ bb0,
                             const float* __restrict__ deg, float* __restrict__ h1, int total)
{
    int i = blockIdx.x * blockDim.x + threadIdx.x;
    if (i >= total) return;
    int c = i & 255;
    int n = i >> 8;
    float norm = 1.0f / fmaxf(deg[n], 1.0f);
    float v = agg[i] + bias1[c] + norm * (buf[i] + bb0[c]);
    h1[i] = v > 0.0f ? v : 0.0f;
}

// ---------------- output combine: out = agg + b2 + norm*(buf+bb1) ---------
__global__ void combine_out_k(const float* __restrict__ agg, const float* __restrict__ bias2,
                              const float* __restrict__ buf, const float* __restrict__ bb1,
                              const float* __restrict__ deg, float* __restrict__ out, int total)
{
    int i = blockIdx.x * blockDim.x + threadIdx.x;
    if (i >= total) return;
    int c = i & 255;
    int n = i >> 8;
    float norm = 1.0f / fmaxf(deg[n], 1.0f);
    out[i] = agg[i] + bias2[c] + norm * (buf[i] + bb1[c]);
}

// ---------------------------------------------------------------------------
extern "C" void kernel_launch(void* const* d_in, const int* in_sizes, int n_in,
                              void* d_out, int out_size, void* d_ws, size_t ws_size,
                              hipStream_t stream)
{
    const float* inputs  = (const float*)d_in[0];
    const int*   src     = (const int*)  d_in[1];
    const int*   dst     = (const int*)  d_in[2];
    const float* fc1_w   = (const float*)d_in[3];
    const float* attn_l1 = (const float*)d_in[4];
    const float* attn_r1 = (const float*)d_in[5];
    const float* bias1   = (const float*)d_in[6];
    const float* fc2_w   = (const float*)d_in[7];
    const float* attn_l2 = (const float*)d_in[8];
    const float* attn_r2 = (const float*)d_in[9];
    const float* bias2   = (const float*)d_in[10];
    const float* wb0     = (const float*)d_in[11];
    const float* bb0     = (const float*)d_in[12];
    const float* wb1     = (const float*)d_in[13];
    const float* bb1     = (const float*)d_in[14];
    float* out = (float*)d_out;

    const int F  = 256;
    const int N  = in_sizes[0] / F;     // 20000
    const int E  = in_sizes[1];         // 320000
    const int H1 = 8, D1 = 32;

    // -------- workspace partition (fp32 elements) --------
    float* ws   = (float*)d_ws;
    float* feat = ws;                ws += (size_t)N * F;   // feat1 then feat2
    float* buf  = ws;                ws += (size_t)N * F;   // buffer-linear outputs
    float* agg  = ws;                ws += (size_t)N * F;   // GAT aggregation
    float* h1   = ws;                ws += (size_t)N * F;
    float* el   = ws;                ws += (size_t)N * H1;
    float* er   = ws;                ws += (size_t)N * H1;
    float* m    = ws;                ws += (size_t)N * H1;
    float* s    = ws;                ws += (size_t)N * H1;
    float* exb  = ws;                ws += (size_t)E * H1;
    float* deg  = ws;                ws += (size_t)N;

    const int TB = 256;
    auto cdiv = [](int a, int b) { return (a + b - 1) / b; };

    const int waves    = (N / 16) * 4;              // 16x64 strip per wave
    const int gemmGrid = cdiv(waves, TB / 32);

    // ================= init =================
    fill_f32<<<cdiv(N, TB), TB, 0, stream>>>(deg, 0.0f, N);
    fill_f32<<<cdiv(N * H1, TB), TB, 0, stream>>>(m, -INFINITY, N * H1);
    fill_f32<<<cdiv(N * H1, TB), TB, 0, stream>>>(s, 0.0f, N * H1);
    fill_f32<<<cdiv(N * F, TB), TB, 0, stream>>>(agg, 0.0f, N * F);
    degree_k<<<cdiv(E, TB), TB, 0, stream>>>(dst, deg, E);

    // ================= layer 1 =================
    gemm_wmma_f32<<<gemmGrid, TB, 0, stream>>>(inputs, F, fc1_w, F, feat, F, N, F, 0);
    gemm_wmma_f32<<<gemmGrid, TB, 0, stream>>>(inputs, F, wb0,   F, buf,  F, N, F, 0);

    attn_scores<<<cdiv(N * H1, TB / 32), TB, 0, stream>>>(feat, attn_l1, attn_r1, el, er, N * H1, H1, D1);
    edge_max_k   <<<cdiv(E * H1, TB), TB, 0, stream>>>(src, dst, el, er, m, E, H1);
    edge_expsum_k<<<cdiv(E * H1, TB), TB, 0, stream>>>(src, dst, el, er, m, s, exb, E, H1);
    edge_agg_k   <<<cdiv(E * (F / 4), TB), TB, 0, stream>>>(src, dst, exb, s, feat, agg, E * (F / 4), H1, D1, F);

    combine_h1_k<<<cdiv(N * F, TB), TB, 0, stream>>>(agg, bias1, buf, bb0, deg, h1, N * F);

    // ================= layer 2 (H=1, D=256) =================
    fill_f32<<<cdiv(N, TB), TB, 0, stream>>>(m, -INFINITY, N);
    fill_f32<<<cdiv(N, TB), TB, 0, stream>>>(s, 0.0f, N);
    fill_f32<<<cdiv(N * F, TB), TB, 0, stream>>>(agg, 0.0f, N * F);

    gemm_wmma_f32<<<gemmGrid, TB, 0, stream>>>(h1, F, fc2_w, F, feat, F, N, F, 0);
    // buf1 = cat(inputs,h1) @ wb1^T  ==  inputs@wb1[:, :256]^T + h1@wb1[:, 256:]^T
    gemm_wmma_f32<<<gemmGrid, TB, 0, stream>>>(inputs, F, wb1,     2 * F, buf, F, N, F, 0);
    gemm_wmma_f32<<<gemmGrid, TB, 0, stream>>>(h1,     F, wb1 + F, 2 * F, buf, F, N, F, 1);

    attn_scores<<<cdiv(N, TB / 32), TB, 0, stream>>>(feat, attn_l2, attn_r2, el, er, N, 1, F);
    edge_max_k   <<<cdiv(E, TB), TB, 0, stream>>>(src, dst, el, er, m, E, 1);
    edge_expsum_k<<<cdiv(E, TB), TB, 0, stream>>>(src, dst, el, er, m, s, exb, E, 1);
    edge_agg_k   <<<cdiv(E * (F / 4), TB), TB, 0, stream>>>(src, dst, exb, s, feat, agg, E * (F / 4), 1, F, F);

    combine_out_k<<<cdiv(N * F, TB), TB, 0, stream>>>(agg, bias2, buf, bb1, deg, out, N * F);
}